// UNetWatershed_67087389164245
// MI455X (gfx1250) — compile-verified
//
#include <hip/hip_runtime.h>

// One-hot expansion: mask [1,1,256,256,48] f32 integer labels (0..40)
//   -> out [1,40,256,256,48] f32, out[n-1][p] = (mask[p] == n)
//
// Pure streaming-store problem: 12.6 MB in, 503 MB out. Roofline floor at
// 23.3 TB/s is ~22 us. Strategy: each lane loads one float4 of mask,
// derives 4 integer labels, and emits 40 non-temporal B128 stores (one per
// label channel). 32 lanes * 16B = 512B contiguous per store instruction.
// NT hint keeps the 503 MB write stream from thrashing the 192 MB L2.

#define VOL   (256 * 256 * 48)   // spatial elements = 3,145,728 (divisible by 4)
#define NLAB  40

typedef __attribute__((ext_vector_type(4))) float fvec4;

__global__ __launch_bounds__(256)
void onehot40_kernel(const float* __restrict__ mask,
                     float* __restrict__ out,
                     int nvec /* = VOL/4 */) {
    int i = blockIdx.x * blockDim.x + threadIdx.x;
    if (i >= nvec) return;

    // Load 4 labels (exact small non-negative ints stored as f32).
    const fvec4 m = ((const fvec4*)mask)[i];
    const int l0 = (int)m.x;
    const int l1 = (int)m.y;
    const int l2 = (int)m.z;
    const int l3 = (int)m.w;

    fvec4* __restrict__ o = (fvec4*)out + i;
    const int chan_stride = VOL / 4;  // in float4 units

    #pragma unroll
    for (int n = 1; n <= NLAB; ++n) {
        fvec4 v;
        v.x = (l0 == n) ? 1.0f : 0.0f;
        v.y = (l1 == n) ? 1.0f : 0.0f;
        v.z = (l2 == n) ? 1.0f : 0.0f;
        v.w = (l3 == n) ? 1.0f : 0.0f;
        // Streaming write-once store: global_store_b128 with TH=NT.
        __builtin_nontemporal_store(v, o);
        o += chan_stride;
    }
}

extern "C" void kernel_launch(void* const* d_in, const int* in_sizes, int n_in,
                              void* d_out, int out_size, void* d_ws, size_t ws_size,
                              hipStream_t stream) {
    (void)n_in; (void)d_ws; (void)ws_size; (void)out_size;

    const float* mask = (const float*)d_in[0];  // [1,1,256,256,48] f32
    // d_in[1] is n_labels (int scalar) == 40, static per reference.
    float* out = (float*)d_out;                 // [1,40,256,256,48] f32

    const int nvec = in_sizes[0] / 4;           // 786,432 float4 units
    const int threads = 256;                    // 8 wave32 per block
    const int blocks = (nvec + threads - 1) / threads;

    onehot40_kernel<<<blocks, threads, 0, stream>>>(mask, out, nvec);
}